// GraphAttentionLayer_9174050144909
// MI455X (gfx1250) — compile-verified
//
#include <hip/hip_runtime.h>
#include <hip/hip_bf16.h>

#define NN   2048
#define FIN  256
#define FOUT 256
#define EE   64
#define GAT_ALPHA 0.2f
#define GAT_NEG_INF -9000000000000000.0f
#define SROW 2052   // padded LDS row stride (floats): lane m -> bank 4m, conflict-free

typedef float v2f __attribute__((ext_vector_type(2)));
typedef float v8f __attribute__((ext_vector_type(8)));

// ---------------------------------------------------------------------------
// Kernel A: h = input @ W   (2048x256 @ 256x256, f32 WMMA 16x16x4)
// One 16x16 output tile per wave. 8 waves/block, 256 blocks -> 2048 tiles.
// ---------------------------------------------------------------------------
__global__ void __launch_bounds__(256)
gat_hproj(const float* __restrict__ input,
          const float* __restrict__ W,
          float* __restrict__ h) {
  const int wid  = threadIdx.x >> 5;
  const int lane = threadIdx.x & 31;
  const int tile = blockIdx.x * 8 + wid;       // 0..2047
  const int i0   = (tile >> 4) * 16;           // M tile
  const int n0   = (tile & 15) * 16;           // N tile
  const int l16  = lane & 15;
  const int hi   = lane >> 4;                  // lanes 16-31 carry K+2
  const int ko   = 2 * hi;

  v8f c = {};
  for (int k = 0; k < FIN; k += 4) {
    v2f a, b;
    a[0] = input[(size_t)(i0 + l16) * FIN + k + ko];
    a[1] = input[(size_t)(i0 + l16) * FIN + k + ko + 1];
    b[0] = W[(size_t)(k + ko)     * FOUT + n0 + l16];
    b[1] = W[(size_t)(k + ko + 1) * FOUT + n0 + l16];
    c = __builtin_amdgcn_wmma_f32_16x16x4_f32(false, a, false, b, (short)0, c,
                                              false, false);
  }
#pragma unroll
  for (int r = 0; r < 8; ++r) {
    h[(size_t)(i0 + r + 8 * hi) * FOUT + n0 + l16] = c[r];
  }
}

// ---------------------------------------------------------------------------
// Kernel B: s_src = h@a1, s_dst = h@a2, v_e = W_e @ (W @ a3)
// blocks 0-7: one row per thread. block 8: v_e via LDS temp.
// ---------------------------------------------------------------------------
__global__ void __launch_bounds__(256)
gat_scalars(const float* __restrict__ h,
            const float* __restrict__ a,
            const float* __restrict__ W,
            const float* __restrict__ W_e,
            float* __restrict__ s_src,
            float* __restrict__ s_dst,
            float* __restrict__ v_e) {
  if (blockIdx.x < 8) {
    const int i = blockIdx.x * 256 + threadIdx.x;
    const float* hr = h + (size_t)i * FOUT;
    float s1 = 0.f, s2 = 0.f;
    for (int n = 0; n < FOUT; ++n) {
      const float hv = hr[n];
      s1 += hv * a[n];
      s2 += hv * a[FOUT + n];
    }
    s_src[i] = s1;
    s_dst[i] = s2;
  } else {
    __shared__ float t[FIN];                   // t = W @ a3
    const int k = threadIdx.x;
    float acc = 0.f;
    for (int n = 0; n < FOUT; ++n) acc += W[(size_t)k * FOUT + n] * a[2 * FOUT + n];
    t[k] = acc;
    __syncthreads();
    if (k < EE) {
      float ve = 0.f;
      for (int q = 0; q < FIN; ++q) ve += W_e[(size_t)k * FIN + q] * t[q];
      v_e[k] = ve;
    }
  }
}

// ---------------------------------------------------------------------------
// Kernel C: fused scores -> masked softmax -> att@h -> elu for 16 rows/block.
// 512 threads (16 waves). Dynamic LDS: 16*SROW scores + 96 aux floats.
// ---------------------------------------------------------------------------
__global__ void __launch_bounds__(512)
gat_attn(const float* __restrict__ e_feat,
         const int*   __restrict__ adj,
         const float* __restrict__ h,
         const float* __restrict__ s_src,
         const float* __restrict__ s_dst,
         const float* __restrict__ v_e,
         float* __restrict__ out) {
  extern __shared__ float smem[];
  float* Sm     = smem;                        // 16 * SROW score strip
  float* inv_l  = smem + 16 * SROW;            // 16 softmax 1/sum
  float* ve_s   = smem + 16 * SROW + 16;       // 64 (16B aligned)
  float* ssrc_s = smem + 16 * SROW + 80;       // 16

  const int tid = threadIdx.x;
  const int i0  = blockIdx.x * 16;

  if (tid < EE) ve_s[tid] = v_e[tid];
  if (tid < 16) ssrc_s[tid] = s_src[i0 + tid];
  __syncthreads();

  // ---- Phase 1: masked leaky-relu scores into LDS (streams 8MB of e_feat)
  const float4* ve4 = (const float4*)ve_s;
  for (int p = 0; p < 64; ++p) {
    const int d = p * 512 + tid;               // 0..32767 pair index
    const int i = d >> 11;                     // row in tile
    const int j = d & (NN - 1);                // column
    const size_t pair = (size_t)(i0 + i) * NN + j;
    const float4* ef = (const float4*)(e_feat + pair * EE);
    if (p < 63) {
      const int dn = d + 512;
      __builtin_prefetch(e_feat + ((size_t)(i0 + (dn >> 11)) * NN +
                                   (dn & (NN - 1))) * EE, 0, 0);
    }
    float acc = 0.f;
#pragma unroll
    for (int q = 0; q < EE / 4; ++q) {
      const float4 e = ef[q];
      const float4 v = ve4[q];
      acc += e.x * v.x + e.y * v.y + e.z * v.z + e.w * v.w;
    }
    float sc = ssrc_s[i] + s_dst[j] + acc;
    sc = (sc >= 0.f) ? sc : GAT_ALPHA * sc;
    Sm[i * SROW + j] = (adj[pair] > 0) ? sc : GAT_NEG_INF;
  }
  __syncthreads();

  // ---- Phase 2: softmax numerators in place, 1/sum to aux (1 wave / row)
  {
    const int w    = tid >> 5;
    const int lane = tid & 31;
    float* row = Sm + w * SROW;
    float m = GAT_NEG_INF;
    for (int q = lane; q < NN; q += 32) m = fmaxf(m, row[q]);
#pragma unroll
    for (int o = 16; o >= 1; o >>= 1) m = fmaxf(m, __shfl_xor(m, o, 32));
    float l = 0.f;
    for (int q = lane; q < NN; q += 32) {
      const float pe = __expf(row[q] - m);
      row[q] = pe;
      l += pe;
    }
#pragma unroll
    for (int o = 16; o >= 1; o >>= 1) l += __shfl_xor(l, o, 32);
    if (lane == 0) inv_l[w] = 1.0f / l;
  }
  __syncthreads();

  // ---- Phase 3: out = elu(diag(1/l) * P @ h), f32 WMMA, 1 N-tile per wave
  {
    const int w    = tid >> 5;                 // N tile 0..15
    const int lane = tid & 31;
    const int l16  = lane & 15;
    const int hi   = lane >> 4;
    const int n0   = w * 16;
    const int ko   = 2 * hi;
    v8f c = {};
    for (int k = 0; k < NN; k += 4) {
      v2f a, b;
      a[0] = Sm[l16 * SROW + k + ko];
      a[1] = Sm[l16 * SROW + k + ko + 1];
      b[0] = h[(size_t)(k + ko)     * FOUT + n0 + l16];
      b[1] = h[(size_t)(k + ko + 1) * FOUT + n0 + l16];
      c = __builtin_amdgcn_wmma_f32_16x16x4_f32(false, a, false, b, (short)0, c,
                                                false, false);
    }
#pragma unroll
    for (int r = 0; r < 8; ++r) {
      const int m = r + 8 * hi;
      float val = c[r] * inv_l[m];
      val = (val > 0.f) ? val : expm1f(val);
      out[(size_t)(i0 + m) * FOUT + n0 + l16] = val;
    }
  }
}

// ---------------------------------------------------------------------------
extern "C" void kernel_launch(void* const* d_in, const int* in_sizes, int n_in,
                              void* d_out, int out_size, void* d_ws, size_t ws_size,
                              hipStream_t stream) {
  const float* input  = (const float*)d_in[0];
  const int*   adj    = (const int*)  d_in[1];
  const float* e_feat = (const float*)d_in[2];
  const float* W_e    = (const float*)d_in[3];
  const float* W      = (const float*)d_in[4];
  const float* a      = (const float*)d_in[5];
  float* out = (float*)d_out;

  float* h     = (float*)d_ws;                  // 2048*256 f32 = 2 MB
  float* s_src = h + (size_t)NN * FOUT;         // 2048
  float* s_dst = s_src + NN;                    // 2048
  float* v_e   = s_dst + NN;                    // 64

  gat_hproj<<<256, 256, 0, stream>>>(input, W, h);
  gat_scalars<<<9, 256, 0, stream>>>(h, a, W, W_e, s_src, s_dst, v_e);

  const size_t smem_bytes = (size_t)(16 * SROW + 96) * sizeof(float); // ~131 KB
  gat_attn<<<NN / 16, 512, smem_bytes, stream>>>(e_feat, adj, h,
                                                 s_src, s_dst, v_e, out);
}